// Resample3d_6554120093741
// MI455X (gfx1250) — compile-verified
//
#include <hip/hip_runtime.h>

// Resample3d (trilinear backward warp), f32.
// Shapes fixed by setup_inputs(): input (2,64,8,128,192), flow (2,3,8,128,192).
// Memory-bound: ~206 MB min traffic -> ~9us floor @ 23.3 TB/s.
// Strategy: 1 thread per spatial voxel (lanes contiguous in W for coalescing),
// hoist the 8 corner offsets + 8 weights out of the 64-channel loop,
// unroll channels 4x for load-latency hiding, non-temporal output stores so
// the ~100MB input stays resident in the 192MB L2.

constexpr int Bn = 2;
constexpr int Cn = 64;
constexpr int Dn = 8;
constexpr int Hn = 128;
constexpr int Wn = 192;
constexpr int SP   = Dn * Hn * Wn;   // 196608 voxels per (b, c) slice
constexpr int NVOX = Bn * SP;        // 393216 spatial threads total

__device__ __forceinline__ int clampi(int v, int hi) {
    v = v < 0 ? 0 : v;
    return v > hi ? hi : v;
}

__global__ __launch_bounds__(256) void resample3d_trilinear(
    const float* __restrict__ inp,   // (B, C, D, H, W)
    const float* __restrict__ flow,  // (B, 3, D, H, W)  channels: dx, dy, dz
    float* __restrict__ out)         // (B, C, D, H, W)
{
    const int tid = blockIdx.x * blockDim.x + threadIdx.x;
    if (tid >= NVOX) return;

    const int b = tid / SP;
    const int s = tid - b * SP;              // z*H*W + y*W + x
    const int z = s / (Hn * Wn);
    const int r = s - z * (Hn * Wn);
    const int y = r / Wn;
    const int x = r - y * Wn;

    // Flow is channels-first: coalesced b32 loads per plane.
    const float* fb = flow + (size_t)b * 3 * SP + s;
    const float px = (float)x + fb[0 * SP];
    const float py = (float)y + fb[1 * SP];
    const float pz = (float)z + fb[2 * SP];

    const float fxf = floorf(px);
    const float fyf = floorf(py);
    const float fzf = floorf(pz);
    const float fx = px - fxf;
    const float fy = py - fyf;
    const float fz = pz - fzf;

    const int x0 = clampi((int)fxf,     Wn - 1);
    const int x1 = clampi((int)fxf + 1, Wn - 1);
    const int y0 = clampi((int)fyf,     Hn - 1);
    const int y1 = clampi((int)fyf + 1, Hn - 1);
    const int z0 = clampi((int)fzf,     Dn - 1);
    const int z1 = clampi((int)fzf + 1, Dn - 1);

    const float wx0 = 1.0f - fx, wx1 = fx;
    const float wy0 = 1.0f - fy, wy1 = fy;
    const float wz0 = 1.0f - fz, wz1 = fz;

    // 8 combined trilinear weights (match reference's product order).
    const float w000 = wz0 * wy0 * wx0;
    const float w001 = wz0 * wy0 * wx1;
    const float w010 = wz0 * wy1 * wx0;
    const float w011 = wz0 * wy1 * wx1;
    const float w100 = wz1 * wy0 * wx0;
    const float w101 = wz1 * wy0 * wx1;
    const float w110 = wz1 * wy1 * wx0;
    const float w111 = wz1 * wy1 * wx1;

    // 8 corner element-offsets within one (D,H,W) slice; invariant over C.
    const int p00 = (z0 * Hn + y0) * Wn;
    const int p01 = (z0 * Hn + y1) * Wn;
    const int p10 = (z1 * Hn + y0) * Wn;
    const int p11 = (z1 * Hn + y1) * Wn;
    const int o000 = p00 + x0, o001 = p00 + x1;
    const int o010 = p01 + x0, o011 = p01 + x1;
    const int o100 = p10 + x0, o101 = p10 + x1;
    const int o110 = p11 + x0, o111 = p11 + x1;

    const float* __restrict__ pb = inp + (size_t)b * Cn * SP;
    float* __restrict__ ob = out + (size_t)b * Cn * SP + s;

    // 64 channels: 8 gathers + 8 FMA each. Unroll 4x -> 32 loads in flight.
    // Output is streamed with non-temporal stores (keep input in L2).
#pragma unroll 4
    for (int c = 0; c < Cn; ++c) {
        const float* __restrict__ p = pb + (size_t)c * SP;
        const float v000 = p[o000];
        const float v001 = p[o001];
        const float v010 = p[o010];
        const float v011 = p[o011];
        const float v100 = p[o100];
        const float v101 = p[o101];
        const float v110 = p[o110];
        const float v111 = p[o111];

        float acc;
        acc = v000 * w000;
        acc = fmaf(v001, w001, acc);
        acc = fmaf(v010, w010, acc);
        acc = fmaf(v011, w011, acc);
        acc = fmaf(v100, w100, acc);
        acc = fmaf(v101, w101, acc);
        acc = fmaf(v110, w110, acc);
        acc = fmaf(v111, w111, acc);

        __builtin_nontemporal_store(acc, ob + (size_t)c * SP);
    }
}

extern "C" void kernel_launch(void* const* d_in, const int* in_sizes, int n_in,
                              void* d_out, int out_size, void* d_ws, size_t ws_size,
                              hipStream_t stream) {
    (void)in_sizes; (void)n_in; (void)out_size; (void)d_ws; (void)ws_size;
    const float* inp  = (const float*)d_in[0];
    const float* flow = (const float*)d_in[1];
    float* out        = (float*)d_out;

    constexpr int kBlock = 256;                       // 8 wave32 waves / block
    constexpr int kGrid  = (NVOX + kBlock - 1) / kBlock;  // 1536 blocks
    resample3d_trilinear<<<kGrid, kBlock, 0, stream>>>(inp, flow, out);
}